// AxialAttentionWithPosition3D_51316269252944
// MI455X (gfx1250) — compile-verified
//
#include <hip/hip_runtime.h>
#include <math.h>

typedef float v2f __attribute__((ext_vector_type(2)));
typedef float v8f __attribute__((ext_vector_type(8)));

#define KDIM   56
#define CIN    64
#define OCH    128
#define BTOT   1024
#define NFLAT  57344          /* BTOT*KDIM */
#define EPSBN  1e-5f

/* workspace layout (float offsets) */
#define OFF_QKV      ((size_t)0)
#define QKV_ELEMS    ((size_t)OCH * NFLAT)            /* 7,340,032 */
#define OFF_SO       (OFF_QKV + QKV_ELEMS)
#define OFF_PSS_SUM  (OFF_SO + QKV_ELEMS)             /* 24*1024 */
#define OFF_PSS_SQ   (OFF_PSS_SUM + 24*1024)
#define OFF_POUT_SUM (OFF_PSS_SQ + 24*1024)           /* 128*1024 */
#define OFF_POUT_SQ  (OFF_POUT_SUM + 128*1024)
#define OFF_SCQ      (OFF_POUT_SQ + 128*1024)
#define OFF_SHQ      (OFF_SCQ + 128)
#define OFF_SCS      (OFF_SHQ + 128)
#define OFF_SHS      (OFF_SCS + 32)
#define OFF_SCO      (OFF_SHS + 32)
#define OFF_SHO      (OFF_SCO + 128)
/* total ~14,991,936 floats ~= 60 MB */

template<int NT>
__device__ inline float block_reduce(float v, float* sm) {
  int tid = threadIdx.x;
  sm[tid] = v; __syncthreads();
  for (int s = NT / 2; s > 0; s >>= 1) {
    if (tid < s) sm[tid] += sm[tid + s];
    __syncthreads();
  }
  float r = sm[0]; __syncthreads();
  return r;
}

/* ---------------- k1: qkv = W(128x64) @ X(64x57344), fp32 WMMA ---------- */
/* qkv layout: [o][a][kk][wd] == [o][n], n=(a*56+kk)*32+wd (contiguous wd). */
__global__ __launch_bounds__(256) void k1_qkv_gemm(
    const float* __restrict__ x, const float* __restrict__ w,
    float* __restrict__ ws) {
  float* qkv = ws + OFF_QKV;
  int lane = threadIdx.x & 31;
  int wave = threadIdx.x >> 5;
  int tile = blockIdx.x * 8 + wave;      /* 28672 tiles */
  int mt = tile & 7;                     /* 8 o-tiles */
  int nt = tile >> 3;                    /* 3584 n-tiles */
  int o0 = mt * 16;
  int n0 = nt * 16;
  int a   = n0 / (KDIM * 32);
  int r1  = n0 % (KDIM * 32);
  int kk  = r1 / 32;
  int wd0 = r1 % 32;                     /* 0 or 16 -> 16 cols contiguous */
  int col = lane & 15;
  int kh  = lane >> 4;
  v8f acc = {0.f,0.f,0.f,0.f,0.f,0.f,0.f,0.f};
  const float* wrow = w + (o0 + col) * CIN;
  for (int c0 = 0; c0 < CIN; c0 += 4) {
    int ca = c0 + 2 * kh;
    v2f afr, bfr;
    afr.x = wrow[ca];
    afr.y = wrow[ca + 1];
    size_t base = ((size_t)(ca * 32 + a) * KDIM + kk) * 32 + wd0 + col;
    bfr.x = x[base];
    bfr.y = x[base + (size_t)NFLAT];     /* +1 in c: stride 32*56*32 */
    acc = __builtin_amdgcn_wmma_f32_16x16x4_f32(
        false, afr, false, bfr, (short)0, acc, false, false);
  }
  int n = n0 + col;
  for (int r = 0; r < 8; ++r) {
    int o = o0 + r + 8 * kh;
    qkv[(size_t)o * NFLAT + n] = acc[r];
  }
}

/* ---------------- k2: per-channel BN stats for qkv ---------------------- */
__global__ __launch_bounds__(256) void k2_qkv_stats(
    const float* __restrict__ g, const float* __restrict__ b, float* ws) {
  __shared__ float sm[256];
  int o = blockIdx.x, tid = threadIdx.x;
  const float* p = ws + OFF_QKV + (size_t)o * NFLAT;
  float s = 0.f, s2 = 0.f;
  for (int i = tid; i < NFLAT; i += 256) { float v = p[i]; s += v; s2 += v*v; }
  float ts  = block_reduce<256>(s,  sm);
  float ts2 = block_reduce<256>(s2, sm);
  if (tid == 0) {
    float mean = ts / (float)NFLAT;
    float var  = ts2 / (float)NFLAT - mean * mean;
    float inv  = rsqrtf(var + EPSBN);
    float sc   = g[o] * inv;
    ws[OFF_SCQ + o] = sc;
    ws[OFF_SHQ + o] = b[o] - mean * sc;
  }
}

/* ---------------- k3: apply qkv BN in place ----------------------------- */
__global__ __launch_bounds__(256) void k3_apply_qkv(float* ws) {
  size_t idx = (size_t)blockIdx.x * 256 + threadIdx.x;   /* exact grid */
  int o = (int)(idx / NFLAT);
  float* qkv = ws + OFF_QKV;
  qkv[idx] = qkv[idx] * ws[OFF_SCQ + o] + ws[OFF_SHQ + o];
}

/* ---------------- k5: partial sum/sumsq of qk/qr/kr per (b,g) ----------- */
__global__ __launch_bounds__(128) void k5_ss_partial(
    const float* __restrict__ rel, float* ws) {
  __shared__ float qs[4][56];
  __shared__ float ksm[4][56];
  __shared__ float sm[128];
  int g = blockIdx.x & 7, b = blockIdx.x >> 3;
  int wd = b & 31, a = b >> 5;
  int tid = threadIdx.x;
  const float* qkv = ws + OFF_QKV;
  for (int idx = tid; idx < 8 * 56; idx += 128) {
    int cc = idx / 56, i = idx % 56;
    float v = qkv[(size_t)(g*16 + cc) * NFLAT + ((size_t)a*56 + i)*32 + wd];
    if (cc < 4) qs[cc][i] = v; else ksm[cc-4][i] = v;
  }
  __syncthreads();
  float sqk=0.f, sqk2=0.f, sqr=0.f, sqr2=0.f, skr=0.f, skr2=0.f;
  for (int idx = tid; idx < 56 * 56; idx += 128) {
    int i = idx / 56, j = idx % 56;
    int d = i - j + 55;
    float qk = 0.f, qr = 0.f, kr = 0.f;
    for (int c = 0; c < 4; ++c) {
      qk += qs[c][i] * ksm[c][j];
      qr += qs[c][i] * rel[c*111 + d];
      kr += ksm[c][i] * rel[(4+c)*111 + d];   /* pre-transpose; sums invariant */
    }
    sqk += qk; sqk2 += qk*qk;
    sqr += qr; sqr2 += qr*qr;
    skr += kr; skr2 += kr*kr;
  }
  float r0 = block_reduce<128>(sqk,  sm);
  float r1 = block_reduce<128>(sqk2, sm);
  float r2 = block_reduce<128>(sqr,  sm);
  float r3 = block_reduce<128>(sqr2, sm);
  float r4 = block_reduce<128>(skr,  sm);
  float r5 = block_reduce<128>(skr2, sm);
  if (tid == 0) {
    ws[OFF_PSS_SUM + (size_t)(0*8+g)*1024 + b] = r0;
    ws[OFF_PSS_SQ  + (size_t)(0*8+g)*1024 + b] = r1;
    ws[OFF_PSS_SUM + (size_t)(1*8+g)*1024 + b] = r2;
    ws[OFF_PSS_SQ  + (size_t)(1*8+g)*1024 + b] = r3;
    ws[OFF_PSS_SUM + (size_t)(2*8+g)*1024 + b] = r4;
    ws[OFF_PSS_SQ  + (size_t)(2*8+g)*1024 + b] = r5;
  }
}

/* ---------------- k5b: finalize 24-channel sim BN ----------------------- */
__global__ __launch_bounds__(256) void k5b_ss_final(
    const float* __restrict__ g, const float* __restrict__ b, float* ws) {
  __shared__ float sm[256];
  int ch = blockIdx.x, tid = threadIdx.x;
  float s = 0.f, s2 = 0.f;
  for (int i = tid; i < 1024; i += 256) {
    s  += ws[OFF_PSS_SUM + (size_t)ch*1024 + i];
    s2 += ws[OFF_PSS_SQ  + (size_t)ch*1024 + i];
  }
  float ts  = block_reduce<256>(s,  sm);
  float ts2 = block_reduce<256>(s2, sm);
  if (tid == 0) {
    float cnt  = 1024.0f * 56.0f * 56.0f;
    float mean = ts / cnt;
    float var  = ts2 / cnt - mean * mean;
    float inv  = rsqrtf(var + EPSBN);
    float sc   = g[ch] * inv;
    ws[OFF_SCS + ch] = sc;
    ws[OFF_SHS + ch] = b[ch] - mean * sc;
  }
}

/* ---------------- k6: attention core per (b,g) -------------------------- */
__global__ __launch_bounds__(128) void k6_attn(
    const float* __restrict__ rel, float* ws) {
  __shared__ float qs[4][56];
  __shared__ float ksm[4][56];
  __shared__ float vs[8][56];
  __shared__ float sim[64 * 64];
  __shared__ float svs[8][56];
  __shared__ float sves[8][56];
  int g = blockIdx.x & 7, b = blockIdx.x >> 3;
  int wd = b & 31, a = b >> 5;
  int tid = threadIdx.x;
  int lane = tid & 31, wave = tid >> 5;
  int col = lane & 15, kh = lane >> 4;
  const float* qkv = ws + OFF_QKV;
  for (int idx = tid; idx < 16 * 56; idx += 128) {
    int cc = idx / 56, i = idx % 56;
    float v = qkv[(size_t)(g*16 + cc) * NFLAT + ((size_t)a*56 + i)*32 + wd];
    if (cc < 4) qs[cc][i] = v;
    else if (cc < 8) ksm[cc-4][i] = v;
    else vs[cc-8][i] = v;
  }
  for (int idx = tid; idx < 64 * 64; idx += 128) sim[idx] = 0.f;
  __syncthreads();

  float sc_qk = ws[OFF_SCS +      g], sh_qk = ws[OFF_SHS +      g];
  float sc_qr = ws[OFF_SCS +  8 + g], sh_qr = ws[OFF_SHS +  8 + g];
  float sc_kr = ws[OFF_SCS + 16 + g], sh_kr = ws[OFF_SHS + 16 + g];

  /* qk = q^T k via WMMA f32 16x16x4 (K = 4 channels exactly) */
  for (int t = wave; t < 16; t += 4) {
    int i0 = (t >> 2) * 16, j0 = (t & 3) * 16;
    int ia = i0 + col, jb = j0 + col, c = 2 * kh;
    v2f afr, bfr;
    afr.x = (ia < 56) ? qs[c][ia]     : 0.f;
    afr.y = (ia < 56) ? qs[c+1][ia]   : 0.f;
    bfr.x = (jb < 56) ? ksm[c][jb]    : 0.f;
    bfr.y = (jb < 56) ? ksm[c+1][jb]  : 0.f;
    v8f cacc = {0.f,0.f,0.f,0.f,0.f,0.f,0.f,0.f};
    cacc = __builtin_amdgcn_wmma_f32_16x16x4_f32(
        false, afr, false, bfr, (short)0, cacc, false, false);
    for (int r = 0; r < 8; ++r) {
      int i = i0 + r + 8 * kh, j = j0 + col;
      if (i < 56 && j < 56) {
        int dq = i - j + 55, dk = j - i + 55;
        float qr = 0.f, kr = 0.f;
        for (int c2 = 0; c2 < 4; ++c2) {
          qr += qs[c2][i]  * rel[c2*111 + dq];
          kr += ksm[c2][j] * rel[(4+c2)*111 + dk];  /* kr transposed */
        }
        sim[i*64 + j] = sc_qk*cacc[r] + sh_qk + sc_qr*qr + sh_qr
                      + sc_kr*kr + sh_kr;
      }
    }
  }
  __syncthreads();

  /* row softmax over j */
  if (tid < 56) {
    int i = tid;
    float m = -1e30f;
    for (int j = 0; j < 56; ++j) m = fmaxf(m, sim[i*64 + j]);
    float s = 0.f;
    for (int j = 0; j < 56; ++j) {
      float e = __expf(sim[i*64 + j] - m);
      sim[i*64 + j] = e; s += e;
    }
    float inv = 1.f / s;
    for (int j = 0; j < 56; ++j) sim[i*64 + j] *= inv;
  }
  __syncthreads();

  /* sv[i,c] = sum_j sim[i,j] v[c,j] via WMMA, K-loop over j */
  {
    int i0 = wave * 16;
    v8f acc = {0.f,0.f,0.f,0.f,0.f,0.f,0.f,0.f};
    for (int j0 = 0; j0 < 56; j0 += 4) {
      int jA = j0 + 2 * kh;
      v2f afr, bfr;
      afr.x = sim[(i0 + col)*64 + jA];
      afr.y = sim[(i0 + col)*64 + jA + 1];
      bfr.x = (col < 8) ? vs[col][jA]     : 0.f;
      bfr.y = (col < 8) ? vs[col][jA + 1] : 0.f;
      acc = __builtin_amdgcn_wmma_f32_16x16x4_f32(
          false, afr, false, bfr, (short)0, acc, false, false);
    }
    for (int r = 0; r < 8; ++r) {
      int i = i0 + r + 8 * kh;
      if (i < 56 && col < 8) svs[col][i] = acc[r];
    }
  }

  /* sve[c,i] = sum_j sim[i,j] * rel[8+c, i-j+55] */
  for (int idx = tid; idx < 8 * 56; idx += 128) {
    int c = idx / 56, i = idx % 56;
    float s = 0.f;
    for (int j = 0; j < 56; ++j)
      s += sim[i*64 + j] * rel[(8+c)*111 + (i - j + 55)];
    sves[c][i] = s;
  }
  __syncthreads();

  /* store so + per-block out-stat partials (deterministic, no atomics) */
  float* so = ws + OFF_SO;
  for (int idx = tid; idx < 16 * 56; idx += 128) {
    int h = idx / 448, rem = idx % 448, c = rem / 56, i = rem % 56;
    float v = h ? sves[c][i] : svs[c][i];
    int o = (g*8 + c)*2 + h;
    so[(size_t)o * NFLAT + ((size_t)a*56 + i)*32 + wd] = v;
  }
  if (tid < 16) {
    int h = tid >> 3, c = tid & 7;
    float s = 0.f, s2 = 0.f;
    for (int i = 0; i < 56; ++i) {
      float v = h ? sves[c][i] : svs[c][i];
      s += v; s2 += v*v;
    }
    int o = (g*8 + c)*2 + h;
    ws[OFF_POUT_SUM + (size_t)o*1024 + b] = s;
    ws[OFF_POUT_SQ  + (size_t)o*1024 + b] = s2;
  }
}

/* ---------------- k6b: finalize 128-channel out BN ---------------------- */
__global__ __launch_bounds__(256) void k6b_out_final(
    const float* __restrict__ g, const float* __restrict__ b, float* ws) {
  __shared__ float sm[256];
  int ch = blockIdx.x, tid = threadIdx.x;
  float s = 0.f, s2 = 0.f;
  for (int i = tid; i < 1024; i += 256) {
    s  += ws[OFF_POUT_SUM + (size_t)ch*1024 + i];
    s2 += ws[OFF_POUT_SQ  + (size_t)ch*1024 + i];
  }
  float ts  = block_reduce<256>(s,  sm);
  float ts2 = block_reduce<256>(s2, sm);
  if (tid == 0) {
    float cnt  = (float)NFLAT;          /* 1024*56 */
    float mean = ts / cnt;
    float var  = ts2 / cnt - mean * mean;
    float inv  = rsqrtf(var + EPSBN);
    float sc   = g[ch] * inv;
    ws[OFF_SCO + ch] = sc;
    ws[OFF_SHO + ch] = b[ch] - mean * sc;
  }
}

/* ---------------- k7: out BN + pair-sum + final transpose --------------- */
__global__ __launch_bounds__(256) void k7_out(
    const float* __restrict__ ws, float* __restrict__ out) {
  size_t idx = (size_t)blockIdx.x * 256 + threadIdx.x;   /* 3,670,016 exact */
  int oc = (int)(idx / NFLAT);
  int rem = (int)(idx % NFLAT);
  const float* so = ws + OFF_SO;
  int o0 = 2*oc, o1 = 2*oc + 1;
  float v0 = so[(size_t)o0*NFLAT + rem] * ws[OFF_SCO + o0] + ws[OFF_SHO + o0];
  float v1 = so[(size_t)o1*NFLAT + rem] * ws[OFF_SCO + o1] + ws[OFF_SHO + o1];
  out[idx] = v0 + v1;   /* layout (oc,a,kk,wd) == required output order */
}

extern "C" void kernel_launch(void* const* d_in, const int* in_sizes, int n_in,
                              void* d_out, int out_size, void* d_ws, size_t ws_size,
                              hipStream_t stream) {
  (void)in_sizes; (void)n_in; (void)out_size; (void)ws_size;
  const float* x   = (const float*)d_in[0];
  const float* w   = (const float*)d_in[1];
  const float* gq  = (const float*)d_in[2];
  const float* bq  = (const float*)d_in[3];
  const float* gs  = (const float*)d_in[4];
  const float* bs  = (const float*)d_in[5];
  const float* go  = (const float*)d_in[6];
  const float* bo  = (const float*)d_in[7];
  const float* rel = (const float*)d_in[8];
  float* out = (float*)d_out;
  float* ws  = (float*)d_ws;

  k1_qkv_gemm <<<3584, 256, 0, stream>>>(x, w, ws);
  k2_qkv_stats<<<128,  256, 0, stream>>>(gq, bq, ws);
  k3_apply_qkv<<<28672,256, 0, stream>>>(ws);
  k5_ss_partial<<<8192,128, 0, stream>>>(rel, ws);
  k5b_ss_final<<<24,   256, 0, stream>>>(gs, bs, ws);
  k6_attn     <<<8192, 128, 0, stream>>>(rel, ws);
  k6b_out_final<<<128, 256, 0, stream>>>(go, bo, ws);
  k7_out      <<<14336,256, 0, stream>>>(ws, out);
}